// Encoder_29996051595531
// MI455X (gfx1250) — compile-verified
//
#include <hip/hip_runtime.h>

#define N_NODES 50000
#define N_EDGES 1600000
#define IN_FEAT 256
#define OUT_FEAT 64

typedef __attribute__((ext_vector_type(2))) float v2f;
typedef __attribute__((ext_vector_type(8))) float v8f;

// ---------------------------------------------------------------------------
// Kernel 1: edge scatter  neigh[dst] += w * feat[src]
// Two wave32s (64 threads) per edge: esub (= tid>>6 within the 4 wave-pairs... actually
// tid>>6 gives a uniform edge per 64-thread pair; with wave32 each wave of 32
// lanes has uniform (src,dst,w) -> scalar loads, and lanes cover half a row
// with float4 (global_load_b128, fully coalesced). Atomic adds go to the
// L2 fp32 atomic units via unsafeAtomicAdd (global_atomic_add_f32).
// 256 threads = 4 edges in flight, 8 iterations -> 32 edges per block.
// N_EDGES = 50000 * 32 exactly, so no bounds checks.
// ---------------------------------------------------------------------------
__global__ __launch_bounds__(256)
void spmm_scatter(const float4* __restrict__ feat4,
                  const int*    __restrict__ esrc,
                  const int*    __restrict__ edst,
                  const float*  __restrict__ ew,
                  float*        __restrict__ neigh)
{
    const int chunk = threadIdx.x & 63;   // which float4 of the 256-wide row
    const int esub  = threadIdx.x >> 6;   // 0..3
    int e = blockIdx.x * 32 + esub;
#pragma unroll
    for (int it = 0; it < 8; ++it, e += 4) {
        const int   s = esrc[e];
        const int   d = edst[e];
        const float w = ew[e];
        const float4 v = feat4[(size_t)s * (IN_FEAT / 4) + chunk];
        float* nrow = &neigh[(size_t)d * IN_FEAT + chunk * 4];
        unsafeAtomicAdd(nrow + 0, v.x * w);
        unsafeAtomicAdd(nrow + 1, v.y * w);
        unsafeAtomicAdd(nrow + 2, v.z * w);
        unsafeAtomicAdd(nrow + 3, v.w * w);
    }
}

// ---------------------------------------------------------------------------
// Kernel 2: fused  out = relu(feat @ W_self + neigh @ W_neigh)
// fp32 WMMA path: V_WMMA_F32_16X16X4_F32, D = A(16x4) x B(4x16) + C(16x16).
// Block = 128 threads = 4 waves; wave w computes the 16x16 output tile at
// columns [16w, 16w+16); block b covers node rows [16b, 16b+16).
// K loop: 256/4 = 64 steps, 2 WMMAs per step (self + neigh) into one acc.
// 50000 % 16 == 0 -> no tail.
// ---------------------------------------------------------------------------
__global__ __launch_bounds__(128)
void gemm_fused_relu(const float* __restrict__ feat,
                     const float* __restrict__ neigh,
                     const float* __restrict__ Wself,
                     const float* __restrict__ Wneigh,
                     float*       __restrict__ out)
{
    const int wave  = threadIdx.x >> 5;        // N-tile: 0..3
    const int lane  = threadIdx.x & 31;
    const int m0    = blockIdx.x * 16;         // node row base
    const int n0    = wave * 16;               // output column base
    const int mrow  = lane & 15;               // A-matrix row for this lane
    const int khalf = lane >> 4;               // 0: K=0,1   1: K=2,3
    const int ncol  = lane & 15;               // B/D column for this lane

    const float* arow_s = feat  + (size_t)(m0 + mrow) * IN_FEAT;
    const float* arow_n = neigh + (size_t)(m0 + mrow) * IN_FEAT;

    v8f acc = {};
    for (int k = 0; k < IN_FEAT; k += 4) {
        const int ka = k + 2 * khalf;          // this lane's first K index
        // A fragments: two consecutive fp32 -> single b64 load
        v2f a_s = *(const v2f*)(arow_s + ka);
        v2f a_n = *(const v2f*)(arow_n + ka);
        // B fragments: W[ka][n0+ncol], W[ka+1][n0+ncol]
        v2f b_s;
        b_s.x = Wself[(size_t)ka * OUT_FEAT + n0 + ncol];
        b_s.y = Wself[(size_t)(ka + 1) * OUT_FEAT + n0 + ncol];
        v2f b_n;
        b_n.x = Wneigh[(size_t)ka * OUT_FEAT + n0 + ncol];
        b_n.y = Wneigh[(size_t)(ka + 1) * OUT_FEAT + n0 + ncol];

        acc = __builtin_amdgcn_wmma_f32_16x16x4_f32(
            false, a_s, false, b_s, (short)0, acc, false, false);
        acc = __builtin_amdgcn_wmma_f32_16x16x4_f32(
            false, a_n, false, b_n, (short)0, acc, false, false);
    }

    // D layout: VGPR i, lanes 0-15 -> M=i ; lanes 16-31 -> M=i+8 ; N = ncol
#pragma unroll
    for (int i = 0; i < 8; ++i) {
        const int m = i + 8 * khalf;
        float v = acc[i];
        v = v > 0.0f ? v : 0.0f;               // ReLU
        out[(size_t)(m0 + m) * OUT_FEAT + n0 + ncol] = v;
    }
}

// ---------------------------------------------------------------------------
extern "C" void kernel_launch(void* const* d_in, const int* in_sizes, int n_in,
                              void* d_out, int out_size, void* d_ws, size_t ws_size,
                              hipStream_t stream)
{
    const float* feat   = (const float*)d_in[0];
    const int*   esrc   = (const int*)  d_in[1];
    const int*   edst   = (const int*)  d_in[2];
    const float* ew     = (const float*)d_in[3];
    const float* Wself  = (const float*)d_in[4];
    const float* Wneigh = (const float*)d_in[5];
    float*       out    = (float*)d_out;
    float*       neigh  = (float*)d_ws;        // 50000*256 fp32 = 51.2 MB

    // Zero the neighbor accumulator (stream-ordered, graph-capturable).
    hipMemsetAsync(neigh, 0, (size_t)N_NODES * IN_FEAT * sizeof(float), stream);

    // Edge scatter: 1.6M edges / 32 per block = 50000 blocks.
    spmm_scatter<<<N_EDGES / 32, 256, 0, stream>>>(
        (const float4*)feat, esrc, edst, ew, neigh);

    // Fused dual-GEMM + ReLU: 50000/16 = 3125 blocks of 4 waves.
    gemm_fused_relu<<<N_NODES / 16, 128, 0, stream>>>(
        feat, neigh, Wself, Wneigh, out);
}